// FingerNetWrapper_36541581754968
// MI455X (gfx1250) — compile-verified
//
#include <hip/hip_runtime.h>
#include <math.h>

#define BB   32
#define HH   80
#define WW   80
#define HW   (HH*WW)        // 6400
#define FH   (HH*8)         // 640
#define FW   (WW*8)         // 640
#define FHW  (FH*FW)        // 409600
#define KSEL 100
#define KPAD 112            // 7 tiles of 16
#define NTILE 7
#define NMBLK 100           // min/max partial blocks per image (100*4096 = 409600)

#define PI_F      3.14159265358979323846f
#define TWO_PI_F  6.28318530717958647692f
#define DEG2RAD_F ((float)(3.14159265358979323846/180.0))
#define ANGLE_T_F ((float)(3.14159265358979323846/6.0))

typedef __attribute__((ext_vector_type(2))) float v2f;
typedef __attribute__((ext_vector_type(8))) float v8f;

// ---------------- 1) round -> separable 5x5 gaussian (sigma=1.5, zero pad) -> round
__global__ __launch_bounds__(256) void k_blur(const float* __restrict__ seg,
                                              float* __restrict__ cleaned) {
  __shared__ float s0[HW];
  __shared__ float s1[HW];
  const int b = blockIdx.x;
  const int tid = threadIdx.x;
  float g[5]; float gs = 0.f;
#pragma unroll
  for (int d = 0; d < 5; ++d) { float c = (float)(d - 2); g[d] = expf(-(c*c) / 4.5f); gs += g[d]; }
#pragma unroll
  for (int d = 0; d < 5; ++d) g[d] /= gs;

  for (int i = tid; i < HW; i += 256) s0[i] = rintf(seg[b*HW + i]);
  __syncthreads();
  for (int i = tid; i < HW; i += 256) {
    int r = i / WW, c = i % WW;
    float acc = 0.f;
#pragma unroll
    for (int d = 0; d < 5; ++d) { int cc = c + d - 2; if (cc >= 0 && cc < WW) acc += g[d] * s0[r*WW + cc]; }
    s1[i] = acc;
  }
  __syncthreads();
  for (int i = tid; i < HW; i += 256) {
    int r = i / WW, c = i % WW;
    float acc = 0.f;
#pragma unroll
    for (int d = 0; d < 5; ++d) { int rr = r + d - 2; if (rr >= 0 && rr < HH) acc += g[d] * s1[rr*WW + c]; }
    cleaned[b*HW + i] = rintf(acc);
  }
}

// ---------------- 2) argmax over 90 orientation channels (coalesced, channel-major strides)
__global__ __launch_bounds__(256) void k_ori_argmax(const float* __restrict__ ori,
                                                    int* __restrict__ ori_idx) {
  int gid = blockIdx.x * 256 + threadIdx.x;
  if (gid >= BB * HW) return;
  int b = gid / HW, p = gid % HW;
  const float* base = ori + (size_t)b * 90 * HW + p;
  float best = base[0]; int bi = 0;
  for (int ch = 1; ch < 90; ++ch) {
    float v = base[(size_t)ch * HW];
    if (v > best) { best = v; bi = ch; }   // first-max tie-break == jnp.argmax
  }
  ori_idx[gid] = bi;
}

// ---------------- 3) per-image top-100 + attribute argmax + WMMA pairwise NMS
__global__ __launch_bounds__(256) void k_topk_nms(
    const float* __restrict__ mscore, const float* __restrict__ cleaned,
    const float* __restrict__ mori, const float* __restrict__ mxoff,
    const float* __restrict__ myoff,
    float* __restrict__ out_mnt, float* __restrict__ out_keep) {
  __shared__ float s_score[HW];
  __shared__ float s_rv[256];
  __shared__ int   s_ri[256];
  __shared__ int   s_sel[KPAD];
  __shared__ float s_x[KPAD], s_y[KPAD], s_ang[KPAD], s_val[KPAD], s_n[KPAD];
  __shared__ unsigned s_sup[KPAD][4];
  __shared__ unsigned s_keep[4];

  const int b = blockIdx.x;
  const int tid = threadIdx.x;

  for (int i = tid; i < HW; i += 256) s_score[i] = mscore[b*HW + i] * cleaned[b*HW + i];
  __syncthreads();

  // iterative max-extraction == lax.top_k (sorted desc, ties -> lower index)
  for (int k = 0; k < KSEL; ++k) {
    float best = -INFINITY; int bi = HW;
    for (int i = tid; i < HW; i += 256) {
      float v = s_score[i];
      if (v > best || (v == best && i < bi)) { best = v; bi = i; }
    }
    s_rv[tid] = best; s_ri[tid] = bi;
    __syncthreads();
    for (int s = 128; s > 0; s >>= 1) {
      if (tid < s) {
        float v2 = s_rv[tid + s]; int i2 = s_ri[tid + s];
        if (v2 > s_rv[tid] || (v2 == s_rv[tid] && i2 < s_ri[tid])) { s_rv[tid] = v2; s_ri[tid] = i2; }
      }
      __syncthreads();
    }
    if (tid == 0) { int sel = s_ri[0]; s_sel[k] = sel; s_val[k] = s_rv[0]; s_score[sel] = -INFINITY; }
    __syncthreads();
  }

  // per-candidate channel argmaxes + mnt output
  if (tid < KPAD) {
    if (tid < KSEL) {
      int idx = s_sel[tid];
      int r = idx / WW, c = idx % WW;
      const float* ob = mori + (size_t)b * 90 * HW + idx;
      float bv = ob[0]; int ai = 0;
      for (int ch = 1; ch < 90; ++ch) { float v = ob[(size_t)ch * HW]; if (v > bv) { bv = v; ai = ch; } }
      const float* xb = mxoff + (size_t)b * 8 * HW + idx;
      float bx = xb[0]; int xo = 0;
      for (int ch = 1; ch < 8; ++ch) { float v = xb[(size_t)ch * HW]; if (v > bx) { bx = v; xo = ch; } }
      const float* yb = myoff + (size_t)b * 8 * HW + idx;
      float by = yb[0]; int yo = 0;
      for (int ch = 1; ch < 8; ++ch) { float v = yb[(size_t)ch * HW]; if (v > by) { by = v; yo = ch; } }
      float x = (float)c * 8.f + (float)xo;
      float y = (float)r * 8.f + (float)yo;
      float ang = ((float)ai * 2.f - 89.f) * DEG2RAD_F;
      s_x[tid] = x; s_y[tid] = y; s_ang[tid] = ang; s_n[tid] = x*x + y*y;
      float* o = out_mnt + ((size_t)b * KSEL + tid) * 4;
      o[0] = x; o[1] = y; o[2] = ang; o[3] = s_val[tid];
    } else {
      s_x[tid] = 0.f; s_y[tid] = 0.f; s_ang[tid] = 0.f; s_n[tid] = 0.f; s_val[tid] = -INFINITY;
    }
    s_sup[tid][0] = 0u; s_sup[tid][1] = 0u; s_sup[tid][2] = 0u; s_sup[tid][3] = 0u;
  }
  __syncthreads();

  // pairwise cross-term via V_WMMA_F32_16X16X4_F32:
  //   A 16x4: lanes 0-15 hold (k0,k1)=(x,y); lanes 16-31 hold (k2,k3)=(0,0)
  //   B 4x16: mirrored layout; C 16x16: lane=col, VGPR r = row r / r+8
  const int wave = tid >> 5, lane = tid & 31;
  const int m = lane & 15; const bool hi = lane >= 16;
  for (int t = wave; t < NTILE * NTILE; t += 8) {
    int ti = t / NTILE, tj = t % NTILE;
    v2f a, bm;
    a.x  = hi ? 0.f : s_x[ti*16 + m];
    a.y  = hi ? 0.f : s_y[ti*16 + m];
    bm.x = hi ? 0.f : s_x[tj*16 + m];
    bm.y = hi ? 0.f : s_y[tj*16 + m];
    v8f c = {0.f,0.f,0.f,0.f,0.f,0.f,0.f,0.f};
    c = __builtin_amdgcn_wmma_f32_16x16x4_f32(false, a, false, bm, (short)0, c, false, false);
    int j = tj*16 + m;
#pragma unroll
    for (int r = 0; r < 8; ++r) {
      int i = ti*16 + r + (hi ? 8 : 0);
      float d2 = s_n[i] + s_n[j] - 2.f * c[r];
      float dist = sqrtf(fmaxf(d2, 0.f));
      float ad = fabsf(s_ang[i] - s_ang[j]);
      float am = fminf(ad, TWO_PI_F - ad);
      bool sup = (i < KSEL) && (j < KSEL) && (dist < 16.f) && (am < ANGLE_T_F);
      unsigned bal = (unsigned)__ballot(sup);  // [15:0] -> row r, [31:16] -> row r+8
      if (lane == 0) {
        int word = tj >> 1; int sh = (tj & 1) * 16;
        atomicOr(&s_sup[ti*16 + r    ][word], (bal & 0xFFFFu) << sh);
        atomicOr(&s_sup[ti*16 + r + 8][word], (bal >> 16)     << sh);
      }
    }
  }
  __syncthreads();

  // sequential suppression: keep = valid; for i: if keep[i]: keep &= ~(sup[i] & (j>i))
  if (tid == 0) {
    unsigned kp[4] = {0u,0u,0u,0u};
    for (int j = 0; j < KSEL; ++j) if (s_val[j] > 0.1f) kp[j >> 5] |= (1u << (j & 31));
    for (int i = 0; i < KSEL; ++i) {
      if ((kp[i >> 5] >> (i & 31)) & 1u) {
#pragma unroll
        for (int w = 0; w < 4; ++w) {
          int base = w * 32;
          unsigned gt;
          if (base > i)            gt = 0xFFFFFFFFu;
          else if (i >= base + 31) gt = 0u;
          else                     gt = ~((1u << (i - base + 1)) - 1u);
          kp[w] &= ~(s_sup[i][w] & gt);
        }
      }
    }
    s_keep[0] = kp[0]; s_keep[1] = kp[1]; s_keep[2] = kp[2]; s_keep[3] = kp[3];
  }
  __syncthreads();
  if (tid < KSEL)
    out_keep[(size_t)b * KSEL + tid] = ((s_keep[tid >> 5] >> (tid & 31)) & 1u) ? 1.f : 0.f;
}

// ---------------- 4a) per-image masked min/max, partial per 4096-elem block
__global__ __launch_bounds__(256) void k_minmax_partial(const float* __restrict__ enh,
                                                        const float* __restrict__ cleaned,
                                                        float* __restrict__ pmin,
                                                        float* __restrict__ pmax) {
  __shared__ float smin[256], smax[256];
  const int img = blockIdx.x / NMBLK, blk = blockIdx.x % NMBLK;
  const int tid = threadIdx.x;
  const float* eb = enh + (size_t)img * FHW;
  const float* cb = cleaned + (size_t)img * HW;
  float lmin = INFINITY, lmax = -INFINITY;
  int base = blk * 4096;
  for (int it = 0; it < 16; ++it) {
    int e = base + it * 256 + tid;
    if (it + 1 < 16) __builtin_prefetch(eb + e + 256, 0, 1);  // global_prefetch_b8
    int y = e / FW, x = e - y * FW;
    float cu = cb[(y >> 3) * WW + (x >> 3)];
    float v = eb[e] * cu;
    lmin = fminf(lmin, v); lmax = fmaxf(lmax, v);
  }
  smin[tid] = lmin; smax[tid] = lmax;
  __syncthreads();
  for (int s = 128; s > 0; s >>= 1) {
    if (tid < s) { smin[tid] = fminf(smin[tid], smin[tid + s]); smax[tid] = fmaxf(smax[tid], smax[tid + s]); }
    __syncthreads();
  }
  if (tid == 0) { pmin[blockIdx.x] = smin[0]; pmax[blockIdx.x] = smax[0]; }
}

// ---------------- 4b) reduce partials -> emin/emax per image
__global__ __launch_bounds__(128) void k_minmax_final(const float* __restrict__ pmin,
                                                      const float* __restrict__ pmax,
                                                      float* __restrict__ emin,
                                                      float* __restrict__ emax) {
  __shared__ float smin[128], smax[128];
  const int b = blockIdx.x, tid = threadIdx.x;
  float lmin = INFINITY, lmax = -INFINITY;
  for (int i = tid; i < NMBLK; i += 128) {
    lmin = fminf(lmin, pmin[b * NMBLK + i]);
    lmax = fmaxf(lmax, pmax[b * NMBLK + i]);
  }
  smin[tid] = lmin; smax[tid] = lmax;
  __syncthreads();
  for (int s = 64; s > 0; s >>= 1) {
    if (tid < s) { smin[tid] = fminf(smin[tid], smin[tid + s]); smax[tid] = fmaxf(smax[tid], smax[tid + s]); }
    __syncthreads();
  }
  if (tid == 0) { emin[b] = smin[0]; emax[b] = smax[0]; }
}

// ---------------- 5) fused writer: enh_visual, cleaned_up*255, ori_field (8 px/thread, float4 x2)
__global__ __launch_bounds__(256) void k_write(const float* __restrict__ enh,
                                               const float* __restrict__ cleaned,
                                               const int* __restrict__ ori_idx,
                                               const float* __restrict__ emin,
                                               const float* __restrict__ emax,
                                               float* __restrict__ out_enh,
                                               float* __restrict__ out_clean,
                                               float* __restrict__ out_ori) {
  int gid = blockIdx.x * 256 + threadIdx.x;
  if (gid >= BB * FH * WW) return;          // one thread per 8 consecutive x-pixels
  int b = gid / (FH * WW);
  int rem = gid % (FH * WW);
  int y = rem / WW, xc = rem % WW;
  int lp = b * HW + (y >> 3) * WW + xc;
  float cu = cleaned[lp];
  float oi = (float)ori_idx[lp];
  float orif = (oi * 2.f - 89.f) * DEG2RAD_F * cu;
  float cu255 = cu * 255.f;
  float mn = emin[b], mx = emax[b];
  float inv = 255.f / (mx - mn + 1e-8f);
  size_t base = (size_t)b * FHW + (size_t)y * FW + (size_t)xc * 8;
  const float4* ep = reinterpret_cast<const float4*>(enh + base);
  float4 e0 = ep[0], e1 = ep[1];
  float4 o0, o1;
  o0.x = (e0.x * cu - mn) * inv; o0.y = (e0.y * cu - mn) * inv;
  o0.z = (e0.z * cu - mn) * inv; o0.w = (e0.w * cu - mn) * inv;
  o1.x = (e1.x * cu - mn) * inv; o1.y = (e1.y * cu - mn) * inv;
  o1.z = (e1.z * cu - mn) * inv; o1.w = (e1.w * cu - mn) * inv;
  reinterpret_cast<float4*>(out_enh + base)[0] = o0;
  reinterpret_cast<float4*>(out_enh + base)[1] = o1;
  float4 cc = {cu255, cu255, cu255, cu255};
  reinterpret_cast<float4*>(out_clean + base)[0] = cc;
  reinterpret_cast<float4*>(out_clean + base)[1] = cc;
  float4 oo = {orif, orif, orif, orif};
  reinterpret_cast<float4*>(out_ori + base)[0] = oo;
  reinterpret_cast<float4*>(out_ori + base)[1] = oo;
}

extern "C" void kernel_launch(void* const* d_in, const int* in_sizes, int n_in,
                              void* d_out, int out_size, void* d_ws, size_t ws_size,
                              hipStream_t stream) {
  const float* seg    = (const float*)d_in[0];
  const float* mscore = (const float*)d_in[1];
  const float* mori   = (const float*)d_in[2];
  const float* mxoff  = (const float*)d_in[3];
  const float* myoff  = (const float*)d_in[4];
  const float* ori    = (const float*)d_in[5];
  const float* enh    = (const float*)d_in[6];

  float* out       = (float*)d_out;
  float* out_mnt   = out;                                   // [B,100,4]
  float* out_keep  = out + (size_t)BB * KSEL * 4;           // [B,100]
  float* out_enh   = out + (size_t)BB * KSEL * 5;           // [B,640,640]
  float* out_clean = out_enh + (size_t)BB * FHW;            // [B,640,640]
  float* out_ori   = out_clean + (size_t)BB * FHW;          // [B,640,640]

  float* ws      = (float*)d_ws;
  float* cleaned = ws;                                      // B*HW
  int*   ori_idx = (int*)(ws + (size_t)BB * HW);            // B*HW
  float* pmin    = ws + (size_t)2 * BB * HW;                // B*100
  float* pmax    = pmin + BB * NMBLK;                       // B*100
  float* emin    = pmax + BB * NMBLK;                       // B
  float* emax    = emin + BB;                               // B

  k_blur<<<BB, 256, 0, stream>>>(seg, cleaned);
  k_ori_argmax<<<(BB * HW + 255) / 256, 256, 0, stream>>>(ori, ori_idx);
  k_topk_nms<<<BB, 256, 0, stream>>>(mscore, cleaned, mori, mxoff, myoff, out_mnt, out_keep);
  k_minmax_partial<<<BB * NMBLK, 256, 0, stream>>>(enh, cleaned, pmin, pmax);
  k_minmax_final<<<BB, 128, 0, stream>>>(pmin, pmax, emin, emax);
  k_write<<<(BB * FH * WW + 255) / 256, 256, 0, stream>>>(enh, cleaned, ori_idx, emin, emax,
                                                          out_enh, out_clean, out_ori);
}